// MetaModel_25864293056628
// MI455X (gfx1250) — compile-verified
//
#include <hip/hip_runtime.h>

typedef __attribute__((ext_vector_type(2))) float v2f;
typedef __attribute__((ext_vector_type(8))) float v8f;

#define NSAMP   256
#define MPATCH  128
#define TSTEPS  200
#define RSTRIDE 129   // 128 + 1 pad: conflict-free row AND column LDS access

__global__ __launch_bounds__(128)
void sir_meta_kernel(const float* __restrict__ Rg,     // (N, M, M)
                     const float* __restrict__ Tg,     // (N, 3, 3)
                     const float* __restrict__ rho0g,  // (N, M, 3)
                     const float* __restrict__ betag,  // (N, 1)
                     float* __restrict__ outg)         // (N, T, M, 4)
{
    extern __shared__ float smem[];
    float* Rl     = smem;                       // 128*129 floats (normalized R)
    float* rhoP   = Rl   + MPATCH * RSTRIDE;    // 3 planes of 128: rho[:,c] column-major
    float* p_s    = rhoP + MPATCH * 3;          // 128: infection prob per patch
    float* ninf_s = p_s  + MPATCH;              // 128: new_inf per patch
    float* Ts     = ninf_s + MPATCH;            // 9 (+pad): transition matrix

    const int n    = blockIdx.x;
    const int tid  = threadIdx.x;
    const int lane = tid & 31;
    const int wave = tid >> 5;
    const int ncol = lane & 15;     // WMMA B/C/D column (N dim)
    const int top  = lane >> 4;     // 0: K/M-low half, 1: K/M-high half

    // ---------------- setup: load R[n], T[n], rho0[n] ----------------
    {
        const float4* Rg4 = (const float4*)(Rg + (size_t)n * MPATCH * MPATCH);
        for (int i = tid; i < MPATCH * MPATCH / 4; i += 128) {
            float4 v = Rg4[i];            // coalesced 16B per lane
            int idx = i * 4;
            int row = idx >> 7, col = idx & 127;
            float* dst = &Rl[row * RSTRIDE + col];
            dst[0] = v.x; dst[1] = v.y; dst[2] = v.z; dst[3] = v.w;
        }
        if (tid < 9) Ts[tid] = Tg[n * 9 + tid];
        // rho0 is (M,3) row-major -> transpose into column planes
        for (int i = tid; i < MPATCH * 3; i += 128) {
            int j = i / 3, c = i % 3;
            rhoP[c * MPATCH + j] = rho0g[(size_t)n * MPATCH * 3 + i];
        }
    }
    __syncthreads();

    // row-normalize R: thread tid owns row tid (conflict-free: bank (tid+k)%64)
    {
        float s = 0.f;
        for (int k = 0; k < MPATCH; ++k) s += Rl[tid * RSTRIDE + k];
        float inv = 1.0f / s;
        for (int k = 0; k < MPATCH; ++k) Rl[tid * RSTRIDE + k] *= inv;
    }
    __syncthreads();

    // neff[k] = sum_j Rnorm[j,k]  (pop == 1); fold beta: p = (beta/neff) * ninf
    float pk_scale;
    {
        float c = 0.f;
        for (int j = 0; j < MPATCH; ++j) c += Rl[j * RSTRIDE + tid];
        pk_scale = betag[n] / c;
    }

    // Constant WMMA B operand: B[k][c] = T[k][c] for k<3,c<3 else 0  (4x16 f32)
    // f32 B 4x16 layout: lane -> col N=lane&15; lanes0-15 hold K={0,1}, lanes16-31 K={2,3}
    // Unconditional loads (clamped address) + value selects: no exec-mask control flow.
    v2f bmat;
    {
        const int cidx = (ncol < 3) ? ncol : 0;          // always-valid address
        float b0 = Ts[(top ? 2 : 0) * 3 + cidx];         // K = 0 / 2
        float b1 = Ts[1 * 3 + cidx];                     // K = 1 (K=3 row is pad)
        bmat.x = (ncol < 3) ? b0 : 0.0f;
        bmat.y = (ncol < 3 && !top) ? b1 : 0.0f;
    }
    __syncthreads();

    float4* ob = (float4*)outg;
    const size_t obase = (size_t)n * TSTEPS * MPATCH;

    // ---------------- 200 sequential SIR steps ----------------
    for (int t = 0; t < TSTEPS; ++t) {
        // emit prev_rho (reference records state BEFORE the update)
        float r0 = rhoP[0 * MPATCH + tid];
        float r1 = rhoP[1 * MPATCH + tid];
        float r2 = rhoP[2 * MPATCH + tid];
        float S  = 1.0f - (r0 + r1 + r2);
        float4 o; o.x = S; o.y = r0; o.z = r1; o.w = r2;
        ob[obase + (size_t)t * MPATCH + tid] = o;

        // Phase A: ninf[tid] = sum_j rho[j,0] * R[j,tid]  (column access, conflict-free)
        float acc = 0.f;
        const float4* rhoI4 = (const float4*)rhoP;       // plane 0 == rho[:,0]
        for (int j4 = 0; j4 < MPATCH / 4; ++j4) {
            float4 rv = rhoI4[j4];                 // uniform -> LDS broadcast
            int j = j4 * 4;
            acc += rv.x * Rl[(j + 0) * RSTRIDE + tid];
            acc += rv.y * Rl[(j + 1) * RSTRIDE + tid];
            acc += rv.z * Rl[(j + 2) * RSTRIDE + tid];
            acc += rv.w * Rl[(j + 3) * RSTRIDE + tid];
        }
        p_s[tid] = acc * pk_scale;                 // p = beta * ninf / neff
        __syncthreads();

        // Phase B: new_inf[tid] = S * sum_k R[tid,k] * p[k]  (row access, conflict-free)
        float acc2 = 0.f;
        const float4* p4 = (const float4*)p_s;
        for (int k4 = 0; k4 < MPATCH / 4; ++k4) {
            float4 pv = p4[k4];                    // uniform -> LDS broadcast
            const float* rr = &Rl[tid * RSTRIDE + k4 * 4];
            acc2 += pv.x * rr[0];
            acc2 += pv.y * rr[1];
            acc2 += pv.z * rr[2];
            acc2 += pv.w * rr[3];
        }
        ninf_s[tid] = S * acc2;                    // wave-local consumers below

        // Compartment update via WMMA f32 16x16x4:
        //   D(16x16) = A(16x4: rho rows, K-padded) x B(4x16: T, padded) + C(new_inf in col 0)
        // Wave w owns rows [32w, 32w+32): all LDS deps here are same-wave (in-order DS).
        #pragma unroll
        for (int tt = 0; tt < 2; ++tt) {
            const int m0   = wave * 32 + tt * 16;
            const int rowA = m0 + ncol;            // A: lane -> row M=lane&15

            // A operand: unconditional plane reads, select on value only
            float ax  = rhoP[(top ? 2 : 0) * MPATCH + rowA];   // K = 0 / 2
            float ay1 = rhoP[1 * MPATCH + rowA];               // K = 1
            v2f a;
            a.x = ax;
            a.y = top ? 0.0f : ay1;                            // K = 3 row is zero pad

            // C operand: new_inf into column 0. Two aligned b128 LDS loads + selects.
            const float4* nf4 = (const float4*)&ninf_s[m0 + 8 * top];  // 32B aligned
            float4 nlo = nf4[0];
            float4 nhi = nf4[1];
            v8f c;
            c[0] = (ncol == 0) ? nlo.x : 0.0f;
            c[1] = (ncol == 0) ? nlo.y : 0.0f;
            c[2] = (ncol == 0) ? nlo.z : 0.0f;
            c[3] = (ncol == 0) ? nlo.w : 0.0f;
            c[4] = (ncol == 0) ? nhi.x : 0.0f;
            c[5] = (ncol == 0) ? nhi.y : 0.0f;
            c[6] = (ncol == 0) ? nhi.z : 0.0f;
            c[7] = (ncol == 0) ? nhi.w : 0.0f;

            v8f d = __builtin_amdgcn_wmma_f32_16x16x4_f32(
                        /*neg_a=*/false, a, /*neg_b=*/false, bmat,
                        /*c_mod=*/(short)0, c, /*reuse_a=*/false, /*reuse_b=*/false);

            // Clip unpredicated (v_max clamp), then ONE uniform-condition store block:
            // lane ncol<3 writes 8 consecutive rows of its plane as two b128 stores.
            v8f valv;
            #pragma unroll
            for (int v = 0; v < 8; ++v)
                valv[v] = fminf(fmaxf(d[v], 0.0f), 1.0f);

            if (ncol < 3) {
                float4 lo, hi;
                lo.x = valv[0]; lo.y = valv[1]; lo.z = valv[2]; lo.w = valv[3];
                hi.x = valv[4]; hi.y = valv[5]; hi.z = valv[6]; hi.w = valv[7];
                float4* dst = (float4*)&rhoP[ncol * MPATCH + m0 + 8 * top]; // 32B aligned
                dst[0] = lo;
                dst[1] = hi;
            }
        }
        __syncthreads();                           // publish rho for next step
    }
}

extern "C" void kernel_launch(void* const* d_in, const int* in_sizes, int n_in,
                              void* d_out, int out_size, void* d_ws, size_t ws_size,
                              hipStream_t stream) {
    (void)in_sizes; (void)n_in; (void)d_ws; (void)ws_size; (void)out_size;
    const float* R    = (const float*)d_in[0];
    const float* T    = (const float*)d_in[1];
    const float* rho0 = (const float*)d_in[2];
    const float* beta = (const float*)d_in[3];
    float* out = (float*)d_out;

    const size_t shmem = (size_t)(MPATCH * RSTRIDE   // Rl
                                  + MPATCH * 3       // rhoP planes
                                  + MPATCH           // p_s
                                  + MPATCH           // ninf_s
                                  + 16)              // Ts (+pad)
                         * sizeof(float);            // ~68.7 KB < 320 KB/WGP

    sir_meta_kernel<<<NSAMP, 128, shmem, stream>>>(R, T, rho0, beta, out);
}